// ScaledDotProductAttention_7799660609952
// MI455X (gfx1250) — compile-verified
//
#include <hip/hip_runtime.h>

typedef __attribute__((ext_vector_type(2))) float v2f;
typedef __attribute__((ext_vector_type(8))) float v8f;

// Problem sizes (fixed by the reference)
constexpr int B  = 4;
constexpr int H  = 8;
constexpr int L  = 2048;
constexpr int DK = 32;
constexpr int DV = 32;

constexpr int QB     = 16;          // query rows per wave (one WMMA M-tile)
constexpr int WAVES  = 8;           // waves per block (256 threads, wave32)
constexpr int QBLOCK = QB * WAVES;  // 128 query rows per block

__global__ __launch_bounds__(256)
void attn_fused_kernel(const float* __restrict__ Qg,
                       const float* __restrict__ Kg,
                       const float* __restrict__ Vg,
                       const unsigned char* __restrict__ Mg,   // bool mask
                       const float* __restrict__ Rg,           // res_att
                       float* __restrict__ Cout,               // context
                       float* __restrict__ Sout)               // scores
{
    // Per-wave LDS staging tile for the D-layout -> A-layout transpose of P.
    // Stride 18 floats: even (keeps b64 alignment) and dodges bank conflicts.
    __shared__ float Pt[WAVES][QB][18];

    const int tid  = threadIdx.x;
    const int wave = tid >> 5;
    const int lane = tid & 31;
    const int half = lane >> 4;   // hi half-wave selector
    const int l16  = lane & 15;

    const int bh    = blockIdx.x / (L / QBLOCK);
    const int qblk  = blockIdx.x % (L / QBLOCK);
    const int qBase = qblk * QBLOCK + wave * QB;

    const float scale = 0.17677669529663687f; // 1/sqrt(32)

    const float*         Qb = Qg + (size_t)bh * L * DK;
    const float*         Kb = Kg + (size_t)bh * L * DK;
    const float*         Vb = Vg + (size_t)bh * L * DV;
    const float*         Rb = Rg + (size_t)bh * L * L;
    const unsigned char* Mb = Mg + (size_t)bh * L * L;
    float*               Sb = Sout + (size_t)bh * L * L;
    float*               Cb = Cout + (size_t)bh * L * DV;

    // ---- Load the 16xDK Q tile into A-matrix layout registers ----
    // A 16x4 fp32 layout: lanes 0-15 hold M=lane; VGPR j at half hi holds
    // K = 4*c + 2*hi + j  for chunk c.
    v2f Aq[8];
    {
        const float* qrow = Qb + (size_t)(qBase + l16) * DK;
        #pragma unroll
        for (int c = 0; c < 8; ++c) {
            const int k0 = 4 * c + 2 * half;
            Aq[c].x = qrow[k0];
            Aq[c].y = qrow[k0 + 1];
        }
    }

    // Online-softmax state (replicated across the 16 lanes of each half-wave;
    // register r corresponds to query row  m = r + 8*half).
    float rowMax[8], rowSum[8];
    #pragma unroll
    for (int r = 0; r < 8; ++r) { rowMax[r] = -3.0e38f; rowSum[r] = 0.0f; }
    v8f acc0 = {};   // context cols  0..15
    v8f acc1 = {};   // context cols 16..31

    for (int kBase = 0; kBase < L; kBase += 16) {
        // ================= S = (Q K^T) 16x16 tile via 8 WMMAs =================
        v8f S = {};
        #pragma unroll
        for (int c = 0; c < 8; ++c) {
            const int k0 = 4 * c + 2 * half;
            const float* krow = Kb + (size_t)(kBase + l16) * DK + k0;
            v2f Bk;
            Bk.x = krow[0];
            Bk.y = krow[1];
            S = __builtin_amdgcn_wmma_f32_16x16x4_f32(
                    false, Aq[c], false, Bk, (short)0, S, false, false);
        }

        // ===== scale + residual + mask; stream scores out (non-temporal) =====
        float p[8];
        #pragma unroll
        for (int r = 0; r < 8; ++r) {
            const int m = r + 8 * half;                 // C-layout row
            const size_t off = (size_t)(qBase + m) * L + kBase + l16;
            float s = S[r] * scale + __builtin_nontemporal_load(Rb + off);
            const unsigned char msk = __builtin_nontemporal_load(Mb + off);
            s = msk ? -1.0e9f : s;
            __builtin_nontemporal_store(s, Sb + off);
            p[r] = s;
        }

        // ============== online softmax update (16-lane reductions) ===========
        #pragma unroll
        for (int r = 0; r < 8; ++r) {
            float v = p[r];
            v = fmaxf(v, __shfl_xor(v, 1, 32));
            v = fmaxf(v, __shfl_xor(v, 2, 32));
            v = fmaxf(v, __shfl_xor(v, 4, 32));
            v = fmaxf(v, __shfl_xor(v, 8, 32));         // tile row max
            const float nm   = fmaxf(rowMax[r], v);
            const float corr = __expf(rowMax[r] - nm);
            rowMax[r] = nm;

            float e = __expf(p[r] - nm);
            p[r] = e;
            float ts = e;
            ts += __shfl_xor(ts, 1, 32);
            ts += __shfl_xor(ts, 2, 32);
            ts += __shfl_xor(ts, 4, 32);
            ts += __shfl_xor(ts, 8, 32);                // tile row sum
            rowSum[r] = rowSum[r] * corr + ts;
            acc0[r] *= corr;
            acc1[r] *= corr;
        }

        // ====== transpose P (C/D-layout -> A-layout) through per-wave LDS ====
        __syncthreads();                                // WAR vs prior reads
        #pragma unroll
        for (int r = 0; r < 8; ++r)
            Pt[wave][r + 8 * half][l16] = p[r];
        __syncthreads();                                // RAW visibility

        // ================= context += P @ V  (2 N-tiles of 16) ===============
        #pragma unroll
        for (int c = 0; c < 4; ++c) {
            const int k0 = 4 * c + 2 * half;
            v2f Ap;
            Ap.x = Pt[wave][l16][k0];
            Ap.y = Pt[wave][l16][k0 + 1];

            const float* vr0 = Vb + (size_t)(kBase + k0)     * DV;
            const float* vr1 = Vb + (size_t)(kBase + k0 + 1) * DV;
            v2f Bv0, Bv1;
            Bv0.x = vr0[l16];       Bv0.y = vr1[l16];
            Bv1.x = vr0[16 + l16];  Bv1.y = vr1[16 + l16];

            acc0 = __builtin_amdgcn_wmma_f32_16x16x4_f32(
                       false, Ap, false, Bv0, (short)0, acc0, false, false);
            acc1 = __builtin_amdgcn_wmma_f32_16x16x4_f32(
                       false, Ap, false, Bv1, (short)0, acc1, false, false);
        }
    }

    // ============ finalize: divide by softmax denom, write context ===========
    #pragma unroll
    for (int r = 0; r < 8; ++r) {
        const int m = r + 8 * half;
        const float inv = 1.0f / rowSum[r];
        float* crow = Cb + (size_t)(qBase + m) * DV;
        crow[l16]      = acc0[r] * inv;
        crow[16 + l16] = acc1[r] * inv;
    }
}

extern "C" void kernel_launch(void* const* d_in, const int* in_sizes, int n_in,
                              void* d_out, int out_size, void* d_ws, size_t ws_size,
                              hipStream_t stream)
{
    (void)in_sizes; (void)n_in; (void)out_size; (void)d_ws; (void)ws_size;

    const float*         Q    = (const float*)d_in[0];
    const float*         K    = (const float*)d_in[1];
    const float*         V    = (const float*)d_in[2];
    const unsigned char* mask = (const unsigned char*)d_in[3];  // bool (1B)
    const float*         res  = (const float*)d_in[4];

    float* ctx    = (float*)d_out;                               // [B,H,L,DV]
    float* scores = (float*)d_out + (size_t)B * H * L * DV;      // [B,H,L,L]

    const int grid = (B * H) * (L / QBLOCK);                     // 512 blocks
    attn_fused_kernel<<<grid, WAVES * 32, 0, stream>>>(
        Q, K, V, mask, res, ctx, scores);
}